// GNNEncoder_1073741824178
// MI455X (gfx1250) — compile-verified
//
#include <hip/hip_runtime.h>
#include <hip/hip_bf16.h>

typedef __attribute__((ext_vector_type(2))) float v2f;
typedef __attribute__((ext_vector_type(8))) float v8f;

// ---------------------------------------------------------------------------
// Async global->LDS copy of 16 bytes per lane (CDNA5, ASYNCcnt-tracked).
// VDST operand = wave-relative LDS byte offset (low 32 bits of generic ptr).
// ---------------------------------------------------------------------------
__device__ __forceinline__ void async_copy_b128(const float* gsrc, float* ldst) {
  unsigned loff = (unsigned)(unsigned long long)ldst;
  asm volatile("global_load_async_to_lds_b128 %0, %1, off"
               :: "v"(loff), "v"((unsigned long long)gsrc) : "memory");
}
__device__ __forceinline__ void wait_async0() {
  asm volatile("s_wait_asynccnt 0x0" ::: "memory");
}

// ---------------------------------------------------------------------------
// Tiled fp32 WMMA GEMM: C[M,N] = A[M,K] @ B[K,N] (+ bias[N])
// BM=128, BN=64, BK=16. 256 threads = 8 waves in 4(M) x 2(N) grid;
// each wave computes 32 rows x 32 cols = four 16x16 f32 WMMA accumulators.
// A tile staged via async global->LDS b128; B tile staged transposed so every
// fragment is one aligned ds_load_b64.
// Requires: K % 16 == 0, N % 64 == 0 (K in {256,128}, N in {256,128}). M guarded.
// ---------------------------------------------------------------------------
template <bool ADD_BIAS>
__global__ __launch_bounds__(256) void gemm_f32_wmma(
    const float* __restrict__ A, const float* __restrict__ B,
    const float* __restrict__ bias, float* __restrict__ C,
    int M, int N, int K) {
  constexpr int BM = 128, BN = 64, BK = 16;
  __shared__ float sA[BM][BK + 4];   // row stride 80B: 16B-aligned, bank-spread
  __shared__ float sBt[BN][BK + 4];  // transposed B tile: sBt[n][k]

  const int tid  = threadIdx.x;
  const int lane = tid & 31;
  const int wave = tid >> 5;     // 0..7
  const int wm   = wave & 3;     // wave tile row (32 rows each)
  const int wn   = wave >> 2;    // wave tile col (32 cols each)
  const int bm   = blockIdx.x * BM;
  const int bn   = blockIdx.y * BN;

  v8f acc00 = {}, acc01 = {}, acc10 = {}, acc11 = {};

  const int l15   = lane & 15;
  const int khalf = (lane >> 4) * 2;        // 0 or 2 (even -> b64-aligned)
  const int mrow0 = wm * 32 + l15;
  const int mrow1 = mrow0 + 16;
  const int ncol  = wn * 32 + l15;

  // staging coordinates
  const int ar = tid >> 1;                  // A: 0..127
  const int ac = (tid & 1) * 8;             // A: 0 or 8
  const int agr = bm + ar;
  const int br = tid >> 4;                  // B: k row 0..15
  const int bc = (tid & 15) * 4;            // B: n col 0..60

  // pre-zero LDS rows for out-of-range A rows (only last M-block takes this)
  if (agr >= M) {
#pragma unroll
    for (int j = 0; j < 8; ++j) sA[ar][ac + j] = 0.f;
  }

  for (int k0 = 0; k0 < K; k0 += BK) {
    // --- A tile (128x16): two async b128 copies per thread, predicated on M ---
    if (agr < M) {
      const float* ap = A + (size_t)agr * K + k0 + ac;
      async_copy_b128(ap, &sA[ar][ac]);
      async_copy_b128(ap + 4, &sA[ar][ac + 4]);
    }
    // --- B tile (16x64) -> transposed LDS: coalesced global, strided ds ---
    {
      const float* bp = B + (size_t)(k0 + br) * N + bn + bc;
      float b0 = bp[0], b1 = bp[1], b2 = bp[2], b3 = bp[3];
      sBt[bc + 0][br] = b0;
      sBt[bc + 1][br] = b1;
      sBt[bc + 2][br] = b2;
      sBt[bc + 3][br] = b3;
    }
    wait_async0();
    __syncthreads();

#pragma unroll
    for (int kk = 0; kk < BK; kk += 4) {
      const int ko = kk + khalf;
      v2f a0 = *(const v2f*)&sA[mrow0][ko];
      v2f a1 = *(const v2f*)&sA[mrow1][ko];
      v2f b0 = *(const v2f*)&sBt[ncol][ko];
      v2f b1 = *(const v2f*)&sBt[ncol + 16][ko];
      acc00 = __builtin_amdgcn_wmma_f32_16x16x4_f32(false, a0, false, b0,
                                                    (short)0, acc00, false, false);
      acc01 = __builtin_amdgcn_wmma_f32_16x16x4_f32(false, a0, false, b1,
                                                    (short)0, acc01, false, false);
      acc10 = __builtin_amdgcn_wmma_f32_16x16x4_f32(false, a1, false, b0,
                                                    (short)0, acc10, false, false);
      acc11 = __builtin_amdgcn_wmma_f32_16x16x4_f32(false, a1, false, b1,
                                                    (short)0, acc11, false, false);
    }
    __syncthreads();
  }

  // --- epilogue: C VGPR r holds (M=r, N=lane&15) lanes 0-15, (M=8+r) lanes 16-31
  const int gn = bn + wn * 32 + l15;
  float bias0 = 0.f, bias1 = 0.f;
  if (ADD_BIAS) {
    bias0 = bias[gn];
    bias1 = bias[gn + 16];
  }
#pragma unroll
  for (int mi = 0; mi < 2; ++mi) {
    const int gm0 = bm + wm * 32 + mi * 16 + (lane >> 4) * 8;
    const v8f& aN0 = mi ? acc10 : acc00;
    const v8f& aN1 = mi ? acc11 : acc01;
#pragma unroll
    for (int r = 0; r < 8; ++r) {
      const int gm = gm0 + r;
      if (gm < M) {
        C[(size_t)gm * N + gn]      = aN0[r] + bias0;
        C[(size_t)gm * N + gn + 16] = aN1[r] + bias1;
      }
    }
  }
}

// ---------------------------------------------------------------------------
// Small helper kernels
// ---------------------------------------------------------------------------
__global__ void zero_f32(float* __restrict__ p, int n) {
  int i = blockIdx.x * blockDim.x + threadIdx.x;
  if (i < n) p[i] = 0.f;
}

// degree count: atomicAdd(deg[dst], 1)
__global__ void deg_count(const int* __restrict__ ei, float* __restrict__ deg, int E) {
  int i = blockIdx.x * blockDim.x + threadIdx.x;
  if (i < E) unsafeAtomicAdd(&deg[ei[E + i]], 1.0f);
}

// dinv = rsqrt(deg + 1)  (self-loop)
__global__ void deg_finalize(float* __restrict__ deg, int n) {
  int i = blockIdx.x * blockDim.x + threadIdx.x;
  if (i < n) deg[i] = rsqrtf(deg[i] + 1.0f);
}

// agg[i,f] = h[i,f] * dinv[i]^2 + bias[f]   (self-loop contribution + conv bias)
__global__ void agg_init(const float* __restrict__ h, const float* __restrict__ dinv,
                         const float* __restrict__ bias, float* __restrict__ agg, int total) {
  int i = blockIdx.x * blockDim.x + threadIdx.x;
  if (i < total) {
    int row = i >> 7;      // F = 128
    int f   = i & 127;
    float d = dinv[row];
    agg[i] = h[i] * (d * d) + bias[f];
  }
}

// one wave (32 lanes) per edge: agg[dst] += norm * h[src], 128 feats, float4/lane
__global__ __launch_bounds__(256) void edge_scatter(
    const int* __restrict__ ei, const float* __restrict__ h,
    const float* __restrict__ dinv, float* __restrict__ agg, int E) {
  int wave = blockIdx.x * (blockDim.x >> 5) + (threadIdx.x >> 5);
  int lane = threadIdx.x & 31;
  if (wave >= E) return;
  int src = ei[wave];
  int dst = ei[E + wave];
  float norm = dinv[src] * dinv[dst];
  const float4* hp = (const float4*)(h + (size_t)src * 128);
  float4 v = hp[lane];
  float* ap = agg + (size_t)dst * 128 + lane * 4;
  unsafeAtomicAdd(ap + 0, v.x * norm);
  unsafeAtomicAdd(ap + 1, v.y * norm);
  unsafeAtomicAdd(ap + 2, v.z * norm);
  unsafeAtomicAdd(ap + 3, v.w * norm);
}

// per-feature sum / sumsq partials -> fp32 atomics into stats[0:128]=sum, [128:256]=sumsq
__global__ __launch_bounds__(256) void bn_stats(
    const float* __restrict__ x, float* __restrict__ stats, int N) {
  const int f    = threadIdx.x & 127;
  const int half = threadIdx.x >> 7;   // 0/1
  float s = 0.f, ss = 0.f;
  for (int r = blockIdx.x * 2 + half; r < N; r += gridDim.x * 2) {
    float v = x[(size_t)r * 128 + f];
    s  += v;
    ss += v * v;
  }
  __shared__ float red[2][2][128];
  red[0][half][f] = s;
  red[1][half][f] = ss;
  __syncthreads();
  if (half == 0) {
    unsafeAtomicAdd(&stats[f],       red[0][0][f] + red[0][1][f]);
    unsafeAtomicAdd(&stats[128 + f], red[1][0][f] + red[1][1][f]);
  }
}

// scale = gamma * rsqrt(var+eps); shift = beta - mu*scale
__global__ void bn_finalize(const float* __restrict__ stats,
                            const float* __restrict__ gamma,
                            const float* __restrict__ beta,
                            float* __restrict__ ss, float invN) {
  int f = threadIdx.x;   // 128 threads
  float mu  = stats[f] * invN;
  float var = stats[128 + f] * invN - mu * mu;
  float sc  = gamma[f] * rsqrtf(var + 1e-5f);
  ss[f]       = sc;
  ss[128 + f] = beta[f] - mu * sc;
}

template <bool RELU>
__global__ void bn_apply(const float* __restrict__ x, const float* __restrict__ ss,
                         float* __restrict__ y, int total) {
  int i = blockIdx.x * blockDim.x + threadIdx.x;
  if (i < total) {
    int f = i & 127;
    float v = x[i] * ss[f] + ss[128 + f];
    if (RELU) v = fmaxf(v, 0.f);
    y[i] = v;
  }
}

// ---------------------------------------------------------------------------
// Launch
// ---------------------------------------------------------------------------
extern "C" void kernel_launch(void* const* d_in, const int* in_sizes, int n_in,
                              void* d_out, int out_size, void* d_ws, size_t ws_size,
                              hipStream_t stream) {
  const float* e_prev = (const float*)d_in[0];
  const int*   edges  = (const int*)d_in[1];    // int32 per harness convention
  const float* proj_W = (const float*)d_in[2];
  const float* proj_b = (const float*)d_in[3];
  const float* W1     = (const float*)d_in[4];
  const float* b1     = (const float*)d_in[5];
  const float* g1     = (const float*)d_in[6];
  const float* be1    = (const float*)d_in[7];
  const float* W2     = (const float*)d_in[8];
  const float* b2     = (const float*)d_in[9];
  const float* g2     = (const float*)d_in[10];
  const float* be2    = (const float*)d_in[11];

  const int Din  = in_sizes[3];          // 256
  const int Dh   = in_sizes[5];          // 128
  const int Dout = in_sizes[9];          // 128
  const int N = in_sizes[0] / Din;       // 100000
  const int E = in_sizes[1] / 2;         // 1600000

  // workspace layout (floats)
  float* ws    = (float*)d_ws;
  float* x     = ws;                          // [N, 256] (later reused as [N,128])
  float* h     = x + (size_t)N * Din;         // [N, 128]
  float* agg   = h + (size_t)N * Dh;          // [N, 128]
  float* dinv  = agg + (size_t)N * Dh;        // [N]
  float* stats = dinv + N;                    // [256]
  float* ssbuf = stats + 256;                 // [256] scale/shift

  const int TPB = 256;
  const int NF  = N * 128;

  // 1) x = e_prev @ proj_W + proj_b
  gemm_f32_wmma<true><<<dim3((N + 127) / 128, Din / 64), TPB, 0, stream>>>(
      e_prev, proj_W, proj_b, x, N, Din, Din);

  // 2) degrees -> dinv
  zero_f32<<<(N + TPB - 1) / TPB, TPB, 0, stream>>>(dinv, N);
  deg_count<<<(E + TPB - 1) / TPB, TPB, 0, stream>>>(edges, dinv, E);
  deg_finalize<<<(N + TPB - 1) / TPB, TPB, 0, stream>>>(dinv, N);

  // 3) h = x @ W1
  gemm_f32_wmma<false><<<dim3((N + 127) / 128, Dh / 64), TPB, 0, stream>>>(
      x, W1, nullptr, h, N, Dh, Din);

  // 4) agg = self-loop + bias, then edge scatter
  agg_init<<<(NF + TPB - 1) / TPB, TPB, 0, stream>>>(h, dinv, b1, agg, NF);
  edge_scatter<<<(E + 7) / 8, TPB, 0, stream>>>(edges, h, dinv, agg, E);

  // 5) BN1 + ReLU -> x (reused)
  zero_f32<<<1, 256, 0, stream>>>(stats, 256);
  bn_stats<<<512, TPB, 0, stream>>>(agg, stats, N);
  bn_finalize<<<1, 128, 0, stream>>>(stats, g1, be1, ssbuf, 1.0f / (float)N);
  bn_apply<true><<<(NF + TPB - 1) / TPB, TPB, 0, stream>>>(agg, ssbuf, x, NF);

  // 6) h = x @ W2
  gemm_f32_wmma<false><<<dim3((N + 127) / 128, Dout / 64), TPB, 0, stream>>>(
      x, W2, nullptr, h, N, Dout, Dh);

  // 7) agg2
  agg_init<<<(NF + TPB - 1) / TPB, TPB, 0, stream>>>(h, dinv, b2, agg, NF);
  edge_scatter<<<(E + 7) / 8, TPB, 0, stream>>>(edges, h, dinv, agg, E);

  // 8) BN2 -> d_out
  zero_f32<<<1, 256, 0, stream>>>(stats, 256);
  bn_stats<<<512, TPB, 0, stream>>>(agg, stats, N);
  bn_finalize<<<1, 128, 0, stream>>>(stats, g2, be2, ssbuf, 1.0f / (float)N);
  bn_apply<false><<<(NF + TPB - 1) / TPB, TPB, 0, stream>>>(agg, ssbuf, (float*)d_out, NF);
}